// GraphAttentionV2Layer_26680336843462
// MI455X (gfx1250) — compile-verified
//
#include <hip/hip_runtime.h>

// ---------------------------------------------------------------------------
// GraphAttentionV2Layer — MI455X (gfx1250, wave32)
//
// Reference reduces to:
//   out[:,   0:128] = h @ W_l
//   out[:, 128:256] = segment_sum(e, receivers) @ W_r   (softmax over size-1
//                     axis == 1 -> weights identically 1; segment_sum is
//                     linear -> fold the 800K-row GEMM into a 50K-row GEMM)
//   out[:, 256:384] = u
//
// Bandwidth-bound (~590 MB traffic, ~3.3 GFLOP) -> full-precision fp32 WMMA
// (V_WMMA_F32_16X16X4_F32); no reason to quantize on a 23.3 TB/s part.
// ---------------------------------------------------------------------------

typedef __attribute__((ext_vector_type(2))) float v2f;
typedef __attribute__((ext_vector_type(8))) float v8f;

#define NNODES 50000
#define NEDGES 800000
#define DF     128      // feature dim (all of D_IN1/D_IN2/D_H1/D_H2/D_U)
#define DOUT   384      // 128 + 128 + 128

// ---------------------------------------------------------------------------
// Kernel 1: se[node][:] = sum of e[edge][:] over edges with receivers[edge]==node
// One wave32 per node. receivers is sorted -> binary search the segment.
// Lane l accumulates columns [4l, 4l+4) as a float4 (32*4 = 128 cols).
// ---------------------------------------------------------------------------
__device__ __forceinline__ int lower_bound_i32(const int* __restrict__ r,
                                               int n, int key) {
  int lo = 0, hi = n;
  while (lo < hi) {
    int mid = (lo + hi) >> 1;
    if (r[mid] < key) lo = mid + 1; else hi = mid;
  }
  return lo;
}

__global__ __launch_bounds__(256) void segsum_kernel(
    const float* __restrict__ e, const int* __restrict__ receivers,
    float* __restrict__ se)
{
  const int wave = threadIdx.x >> 5;
  const int lane = threadIdx.x & 31;
  const int node = blockIdx.x * 8 + wave;
  if (node >= NNODES) return;

  const int start = lower_bound_i32(receivers, NEDGES, node);
  const int end   = lower_bound_i32(receivers, NEDGES, node + 1);

  float4 acc = make_float4(0.f, 0.f, 0.f, 0.f);
  const int c4 = lane * 4;
  for (int ed = start; ed < end; ++ed) {
    const float4 v = *(const float4*)(e + (size_t)ed * DF + c4);
    acc.x += v.x; acc.y += v.y; acc.z += v.z; acc.w += v.w;
  }
  *(float4*)(se + (size_t)node * DF + c4) = acc;   // zeros for empty segments
}

// ---------------------------------------------------------------------------
// Kernel 2: fused dual GEMM (16x128 tile per wave, fp32 WMMA 16x16x4) + u copy
//
// f32 WMMA fragment layouts (ISA 7.12.2):
//   A 16x4 : lane L, elem j  -> A[L%16][k0 + 2*(L/16) + j]      (float2 load)
//   B 4x16 : lane L, elem j  -> B[k0 + 2*(L/16) + j][n0 + L%16]
//   C/D    : lane L, elem v  -> D[v + 8*(L/16)][n0 + L%16]
// ---------------------------------------------------------------------------
__device__ __forceinline__ void gemm16x128_tile(
    const float* __restrict__ A,     // [NNODES, 128] row-major
    const float* __restrict__ W,     // [128, 128] row-major (K x N)
    float* __restrict__ out,         // [NNODES, 384]
    int colbase, int r0, int lane)
{
  const int half = lane >> 4;        // 0 or 1
  const int l16  = lane & 15;

  const int arow = r0 + l16;
  const float* aptr = A + (size_t)(arow < NNODES ? arow : 0) * DF;

  v8f acc[8];
#pragma unroll
  for (int t = 0; t < 8; ++t) acc[t] = (v8f)(0.0f);

  for (int kk = 0; kk < DF; kk += 4) {
    const int kb = kk + 2 * half;
    const v2f a = *(const v2f*)(aptr + kb);            // contiguous K pair
#pragma unroll
    for (int t = 0; t < 8; ++t) {
      v2f b;
      b.x = W[(size_t)kb       * DF + t * 16 + l16];
      b.y = W[(size_t)(kb + 1) * DF + t * 16 + l16];
      // v_wmma_f32_16x16x4_f32: (neg_a, A, neg_b, B, c_mod, C, reuse_a, reuse_b)
      acc[t] = __builtin_amdgcn_wmma_f32_16x16x4_f32(
          false, a, false, b, (short)0, acc[t], false, false);
    }
  }

  // Wave-uniform tail guard: only the last block can be partial, so the
  // common path is straight-line, unpredicated, coalesced b32 stores
  // (no per-element EXEC save/restore churn).
  float* obase = out + (size_t)(r0 + 8 * half) * DOUT + colbase + l16;
  if (r0 + 16 <= NNODES) {
#pragma unroll
    for (int t = 0; t < 8; ++t) {
#pragma unroll
      for (int v = 0; v < 8; ++v)
        obase[(size_t)v * DOUT + t * 16] = acc[t][v];
    }
  } else if (r0 < NNODES) {            // tail wave (at most one per grid)
#pragma unroll
    for (int t = 0; t < 8; ++t) {
#pragma unroll
      for (int v = 0; v < 8; ++v) {
        const int row = r0 + v + 8 * half;
        if (row < NNODES)
          out[(size_t)row * DOUT + colbase + t * 16 + l16] = acc[t][v];
      }
    }
  }
}

__global__ __launch_bounds__(256) void fused_gemm_concat_kernel(
    const float* __restrict__ h,  const float* __restrict__ se,
    const float* __restrict__ u,  const float* __restrict__ Wl,
    const float* __restrict__ Wr, float* __restrict__ out)
{
  const int lane = threadIdx.x & 31;
  const int wave = threadIdx.x >> 5;
  const int rb   = blockIdx.x * 128;      // 8 waves x 16 rows
  const int r0   = rb + wave * 16;

  // cols [0,128): h @ W_l      cols [128,256): segsum(e) @ W_r
  gemm16x128_tile(h,  Wl, out, 0,   r0, lane);
  gemm16x128_tile(se, Wr, out, 128, r0, lane);

  // cols [256,384): copy u (float4, coalesced); block-uniform tail guard
  if (rb + 128 <= NNODES) {
#pragma unroll 4
    for (int idx = threadIdx.x; idx < 128 * (DF / 4); idx += 256) {
      const int row = rb + (idx >> 5);
      const int c4  = (idx & 31) * 4;
      const float4 val = *(const float4*)(u + (size_t)row * DF + c4);
      *(float4*)(out + (size_t)row * DOUT + 256 + c4) = val;
    }
  } else {
    for (int idx = threadIdx.x; idx < 128 * (DF / 4); idx += 256) {
      const int row = rb + (idx >> 5);
      const int c4  = (idx & 31) * 4;
      if (row < NNODES) {
        const float4 val = *(const float4*)(u + (size_t)row * DF + c4);
        *(float4*)(out + (size_t)row * DOUT + 256 + c4) = val;
      }
    }
  }
}

// ---------------------------------------------------------------------------
extern "C" void kernel_launch(void* const* d_in, const int* in_sizes, int n_in,
                              void* d_out, int out_size, void* d_ws, size_t ws_size,
                              hipStream_t stream) {
  const float* h         = (const float*)d_in[0];
  const float* e         = (const float*)d_in[1];
  const int*   receivers = (const int*)  d_in[2];
  const float* u         = (const float*)d_in[3];
  const float* Wl        = (const float*)d_in[4];
  const float* Wr        = (const float*)d_in[5];
  // d_in[6] (attn_w) is provably dead: softmax over a size-1 axis == 1.
  float* out = (float*)d_out;
  float* se  = (float*)d_ws;              // [NNODES, 128] fp32 = 25.6 MB

  segsum_kernel<<<(NNODES + 7) / 8, 256, 0, stream>>>(e, receivers, se);
  fused_gemm_concat_kernel<<<(NNODES + 127) / 128, 256, 0, stream>>>(
      h, se, u, Wl, Wr, out);
}